// BeijingGeometry_86423331930747
// MI455X (gfx1250) — compile-verified
//
#include <hip/hip_runtime.h>
#include <stdint.h>

// ---------------------------------------------------------------------------
// Problem constants (from the reference)
// ---------------------------------------------------------------------------
#define A_N   96
#define P_N   4
#define DET1  8
#define DET0C 256
#define NZ    32
#define NY    256
#define NX    256
#define T_N   256
#define LAMB  0.01f

#define PI_OVER_A 0.0327249234748936805f  // pi / 96

// LDS map for the forward kernel (floats):
//   [0, 65536)          : z-slice of the volume (256 KB)
//   [65536, 66560)      : 1024 t-partials, layout part[u*4 + tq]
//   [66560, 66816)      : 256 reduced sums S[u] (WMMA path)
#define SLICE_F   (NY * NX)
#define PART_OFF  SLICE_F
#define SUM_OFF   (SLICE_F + 1024)
#define FWD_LDS_F (SLICE_F + 1024 + 256)

// ---------------------------------------------------------------------------
// Feature gates
// ---------------------------------------------------------------------------
#if __has_builtin(__builtin_amdgcn_tensor_load_to_lds) && __has_builtin(__builtin_amdgcn_s_wait_tensorcnt)
#define USE_TDM 1
#else
#define USE_TDM 0
#endif

#if __has_builtin(__builtin_amdgcn_wmma_f32_16x16x4_f32)
#define USE_WMMA_RED 1
#else
#define USE_WMMA_RED 0
#endif

typedef unsigned int v4u __attribute__((ext_vector_type(4)));
typedef int          v4i __attribute__((ext_vector_type(4)));
typedef int          v8i __attribute__((ext_vector_type(8)));
typedef float        v2f __attribute__((ext_vector_type(2)));
typedef float        v8f __attribute__((ext_vector_type(8)));

// ---------------------------------------------------------------------------
// TDM (Tensor Data Mover) helper: one-instruction async DMA global -> LDS.
// Descriptor layout per CDNA5 ISA (D# groups 0..1;
// groups 2/3 zero => 2D tensor). Verified in round-1 asm: lowers to
// `tensor_load_to_lds s[..], s[..]` with all fields in SGPRs.
// ---------------------------------------------------------------------------
__device__ __forceinline__ void tdm_load_2d_to_lds(
    uint32_t lds_off, uint64_t gaddr,
    uint32_t tensor_d0, uint32_t tensor_d1,
    uint32_t tile_d0, uint32_t tile_d1, uint32_t stride0)
{
#if USE_TDM
    v4u g0;
    g0[0] = 1u;                                            // count=1, user D#
    g0[1] = lds_off;                                       // lds_addr (bytes)
    g0[2] = (uint32_t)(gaddr & 0xffffffffull);             // global_addr[31:0]
    g0[3] = ((uint32_t)(gaddr >> 32) & 0x01ffffffu)        // global_addr[56:32]
          | (2u << 30);                                    // type=2 ("image")
    v8i g1;
    g1[0] = (int)(2u << 16);                               // data_size=2 -> 4B
    g1[1] = (int)((tensor_d0 & 0xffffu) << 16);            // tensor_dim0[15:0]
    g1[2] = (int)((tensor_d0 >> 16) | ((tensor_d1 & 0xffffu) << 16));
    g1[3] = (int)((tensor_d1 >> 16) | (tile_d0 << 16));    // tile_dim0
    g1[4] = (int)(tile_d1 & 0xffffu);                      // tile_dim1, tile_dim2=0
    g1[5] = (int)stride0;                                  // tensor_dim0_stride[31:0]
    g1[6] = 0;
    g1[7] = 0;
    v4i g2 = {0, 0, 0, 0};
    v4i g3 = {0, 0, 0, 0};
#if defined(__clang_major__) && (__clang_major__ >= 23)
    v8i g4 = {0, 0, 0, 0, 0, 0, 0, 0};
    __builtin_amdgcn_tensor_load_to_lds(g0, g1, g2, g3, g4, 0);
#else
    __builtin_amdgcn_tensor_load_to_lds(g0, g1, g2, g3, 0);
#endif
#else
    (void)lds_off; (void)gaddr; (void)tensor_d0; (void)tensor_d1;
    (void)tile_d0; (void)tile_d1; (void)stride0;
#endif
}

__device__ __forceinline__ int iclamp(int v, int lo, int hi) {
    return v < lo ? lo : (v > hi ? hi : v);
}

// ---------------------------------------------------------------------------
// Kernel 1: forward projection + residual (writes res_native[a][z][u]).
// One block per (angle, z-slice): whole 256 KB slice staged into LDS by one
// TDM DMA (CDNA5 WGP has 320 KB LDS), then 1024 threads (4 t-quarters x 256
// detector bins) march along rays doing 4-tap bilinear LDS gathers. The
// final tq-reduction S[u] = sum_k part[u][k] is done as 16 exact WMMA row
// sums: D(16x16) = A(16x4) x ones(4x16), C=0 — columns of D replicate S.
// ---------------------------------------------------------------------------
__global__ __launch_bounds__(1024)
void fwd_residual_kernel(const float* __restrict__ vol,
                         const float* __restrict__ p,
                         float* __restrict__ res)
{
    extern __shared__ float lds[];
    const int tid = threadIdx.x;
    const int a = blockIdx.x / NZ;
    const int z = blockIdx.x - a * NZ;

#if USE_TDM
    if (tid < 32) {   // wave 0 issues the DMA (per-wave op, EXEC-independent)
        const uint64_t ga = (uint64_t)(uintptr_t)(vol + (size_t)z * NY * NX);
        const uint32_t lo = (uint32_t)(uintptr_t)(&lds[0]);
        tdm_load_2d_to_lds(lo, ga, 16384u, 4u, 16384u, 4u, 16384u);
        __builtin_amdgcn_s_wait_tensorcnt((short)0);
    }
    __syncthreads();
#else
    {
        const float4* src = (const float4*)(vol + (size_t)z * NY * NX);
        float4* dst = (float4*)lds;
        for (int i = tid; i < (NY * NX) / 4; i += blockDim.x) dst[i] = src[i];
        __syncthreads();
    }
#endif

    const int u  = tid & (DET0C - 1);
    const int tq = tid >> 8;                       // t-quarter 0..3
    const float th = (float)a * PI_OVER_A;
    const float c = cosf(th);
    const float s = sinf(th);
    const float uf = (float)u - 127.5f;
    const float bx = fmaf(-uf, s, 127.5f);         // x = -u*s + t*c + cx
    const float by = fmaf( uf, c, 127.5f);         // y =  u*c + t*s + cy

    float acc = 0.0f;
    const int t0 = tq << 6;
#pragma unroll 4
    for (int tt = 0; tt < 64; ++tt) {
        const float tf = (float)(t0 + tt) - 127.5f;
        const float px = fmaf(tf, c, bx);
        const float py = fmaf(tf, s, by);
        const float xf0 = floorf(px);
        const float yf0 = floorf(py);
        const float fx = px - xf0;
        const float fy = py - yf0;
        const int ix0 = (int)xf0, iy0 = (int)yf0;
        const int ix1 = ix0 + 1,  iy1 = iy0 + 1;
        const float vx0 = (ix0 >= 0 && ix0 < NX) ? 1.0f : 0.0f;
        const float vx1 = (ix1 >= 0 && ix1 < NX) ? 1.0f : 0.0f;
        const float vy0 = (iy0 >= 0 && iy0 < NY) ? 1.0f : 0.0f;
        const float vy1 = (iy1 >= 0 && iy1 < NY) ? 1.0f : 0.0f;
        const int cx0 = iclamp(ix0, 0, NX - 1);
        const int cx1 = iclamp(ix1, 0, NX - 1);
        const int cy0 = iclamp(iy0, 0, NY - 1) * NX;
        const int cy1 = iclamp(iy1, 0, NY - 1) * NX;
        const float s00 = lds[cy0 + cx0];
        const float s01 = lds[cy0 + cx1];
        const float s10 = lds[cy1 + cx0];
        const float s11 = lds[cy1 + cx1];
        const float gx = 1.0f - fx, gy = 1.0f - fy;
        acc = fmaf(gy * gx * (vy0 * vx0), s00, acc);
        acc = fmaf(gy * fx * (vy0 * vx1), s01, acc);
        acc = fmaf(fy * gx * (vy1 * vx0), s10, acc);
        acc = fmaf(fy * fx * (vy1 * vx1), s11, acc);
    }

    lds[PART_OFF + (u << 2) + tq] = acc;           // part[u][tq]
    __syncthreads();

#if USE_WMMA_RED
    // Waves 0..15 each reduce one 16-u tile: D = A(16x4) * ones(4x16) + 0.
    // 32-bit A 16x4 layout (ISA 7.12.2): lanes 0-15 hold K=0,1 (VGPR0/1),
    // lanes 16-31 hold K=2,3 for the same M rows.
    if (tid < 512) {                               // 16 waves, uniform branch
        const int w    = tid >> 5;                 // tile index 0..15
        const int lane = tid & 31;
        const int m    = lane & 15;                // row within tile
        const int kh   = lane >> 4;                // 0 -> K=0,1 ; 1 -> K=2,3
        const float* part = lds + PART_OFF + ((w * 16 + m) << 2) + (kh << 1);
        v2f av; av.x = part[0]; av.y = part[1];
        v2f bv; bv.x = 1.0f;    bv.y = 1.0f;       // ones: layout-independent
        v8f cv = {0.f, 0.f, 0.f, 0.f, 0.f, 0.f, 0.f, 0.f};
        v8f dv = __builtin_amdgcn_wmma_f32_16x16x4_f32(
            false, av, false, bv, (short)0, cv, false, false);
        // Column N=0 lives in lane 0 (M=r) and lane 16 (M=r+8), r=0..7.
        if ((lane & 15) == 0) {
            const int mbase = w * 16 + (lane >> 4) * 8;
#pragma unroll
            for (int r = 0; r < 8; ++r)
                lds[SUM_OFF + mbase + r] = dv[r];
        }
    }
    __syncthreads();
    if (tid < DET0C) {
        const float S = lds[SUM_OFF + tid];
        const int d_i = z & 7, p_i = z >> 3;
        const float pv = p[(size_t)(a * 32 + d_i * 4 + p_i) * DET0C + tid];
        res[((size_t)a * NZ + z) * DET0C + tid] = S - pv;
    }
#else
    if (tid < DET0C) {
        const float* pr = lds + PART_OFF + (tid << 2);
        const float S = (pr[0] + pr[1]) + (pr[2] + pr[3]);
        // res_native[a,z,u] = S[a,z,u] - p[a*32 + (z&7)*4 + (z>>3)][u]
        const int d_i = z & 7, p_i = z >> 3;
        const float pv = p[(size_t)(a * 32 + d_i * 4 + p_i) * DET0C + tid];
        res[((size_t)a * NZ + z) * DET0C + tid] = S - pv;
    }
#endif
}

// ---------------------------------------------------------------------------
// Kernel 2: back projection. One block per 16x16 (y,x) tile; each thread
// keeps 32 z-accumulators in VGPRs. Per angle, the 32 KB residual slab
// res[a][*][*] is double-buffered into LDS via TDM; u-interp weights are
// computed once per angle and amortized over all 32 z (2 LDS reads + 2 FMA).
// ---------------------------------------------------------------------------
__global__ __launch_bounds__(256)
void bwd_kernel(const float* __restrict__ res, float* __restrict__ out)
{
    extern __shared__ float lds[];   // 2 x 8192 floats (double buffer)
    const int tid = threadIdx.x;
    const int x = ((blockIdx.x & 15) << 4) + (tid & 15);
    const int y = ((blockIdx.x >> 4) << 4) + (tid >> 4);
    const float xf = (float)x - 127.5f;
    const float yf = (float)y - 127.5f;

    float acc[NZ];
#pragma unroll
    for (int z = 0; z < NZ; ++z) acc[z] = 0.0f;

#if USE_TDM
    if (tid < 32) {   // preload slab 0 into buffer 0
        tdm_load_2d_to_lds((uint32_t)(uintptr_t)(&lds[0]),
                           (uint64_t)(uintptr_t)res,
                           8192u, 1u, 8192u, 1u, 8192u);
    }
#endif

    for (int a = 0; a < A_N; ++a) {
#if USE_TDM
        if (tid < 32) {
            if (a + 1 < A_N) {  // prefetch next slab into the other buffer
                tdm_load_2d_to_lds((uint32_t)(uintptr_t)(&lds[((a + 1) & 1) << 13]),
                                   (uint64_t)(uintptr_t)(res + (size_t)(a + 1) * NZ * DET0C),
                                   8192u, 1u, 8192u, 1u, 8192u);
                __builtin_amdgcn_s_wait_tensorcnt((short)1);  // slab a done
            } else {
                __builtin_amdgcn_s_wait_tensorcnt((short)0);
            }
        }
        __syncthreads();
        const float* Rb = lds + ((a & 1) << 13);
#else
        __syncthreads();
        for (int i = tid; i < NZ * DET0C; i += 256)
            lds[i] = res[(size_t)a * NZ * DET0C + i];
        __syncthreads();
        const float* Rb = lds;
#endif

        const float th = (float)a * PI_OVER_A;
        const float c = cosf(th);
        const float s = sinf(th);
        const float ub = fmaf(-xf, s, fmaf(yf, c, 127.5f));
        const float u0f = floorf(ub);
        const float fu = ub - u0f;
        const int u0i = (int)u0f;
        const int u1i = u0i + 1;
        const float w0 = (1.0f - fu) * ((u0i >= 0 && u0i < DET0C) ? 1.0f : 0.0f);
        const float w1 = fu          * ((u1i >= 0 && u1i < DET0C) ? 1.0f : 0.0f);
        const int i0 = iclamp(u0i, 0, DET0C - 1);
        const int i1 = iclamp(u1i, 0, DET0C - 1);

#pragma unroll
        for (int z = 0; z < NZ; ++z) {
            const float r0 = Rb[z * DET0C + i0];
            const float r1 = Rb[z * DET0C + i1];
            acc[z] = fmaf(r0, w0, fmaf(r1, w1, acc[z]));
        }
#if USE_TDM
        __syncthreads();   // all reads of this buffer done before it is reused
#endif
    }

#pragma unroll
    for (int z = 0; z < NZ; ++z)
        out[((size_t)z * NY + y) * NX + x] = -LAMB * acc[z];
}

// ---------------------------------------------------------------------------
// Host launcher
// ---------------------------------------------------------------------------
extern "C" void kernel_launch(void* const* d_in, const int* in_sizes, int n_in,
                              void* d_out, int out_size, void* d_ws, size_t ws_size,
                              hipStream_t stream)
{
    const float* x = (const float*)d_in[0];   // (1,1,32,256,256) f32
    const float* p = (const float*)d_in[1];   // (1,1,384,8,256)  f32
    float* out = (float*)d_out;               // (1,1,32,256,256) f32
    float* res = (float*)d_ws;                // A*NZ*DET0 f32 = 3 MB scratch

    const size_t shmem_fwd = (size_t)FWD_LDS_F * sizeof(float);        // ~261 KB
    const size_t shmem_bwd = (size_t)(2 * NZ * DET0C) * sizeof(float); //   64 KB

    // Allow >64KB dynamic LDS (CDNA5 WGP has 320KB). Idempotent, capture-safe.
    (void)hipFuncSetAttribute((const void*)fwd_residual_kernel,
                              hipFuncAttributeMaxDynamicSharedMemorySize,
                              (int)shmem_fwd);
    (void)hipFuncSetAttribute((const void*)bwd_kernel,
                              hipFuncAttributeMaxDynamicSharedMemorySize,
                              (int)shmem_bwd);

    fwd_residual_kernel<<<A_N * NZ, 1024, shmem_fwd, stream>>>(x, p, res);
    bwd_kernel<<<(NY / 16) * (NX / 16), 256, shmem_bwd, stream>>>(res, out);
}